// Decoder_54202487275648
// MI455X (gfx1250) — compile-verified
//
#include <hip/hip_runtime.h>
#include <math.h>

// ---------------------------------------------------------------------------
// Problem constants (from the reference)
// ---------------------------------------------------------------------------
#define N_NODES 50000
#define BATCH   512
#define DIM     32
#define NSAMP   1001          // int(10.0/0.01)+1
#define N_TILES (N_NODES / 16)   // 3125

typedef __attribute__((ext_vector_type(2))) float v2f;
typedef __attribute__((ext_vector_type(4))) float v4f;
typedef __attribute__((ext_vector_type(8))) float v8f;

// Numerically-stable softplus == jnp.logaddexp(0,-x)+x.
// exp(-|x|) in (0,1] so log1p(t) ~= __logf(1+t): one v_exp_f32 + one
// v_log_f32 instead of the ~15-op OCML log1p polynomial (abs err < 1e-7).
__device__ __forceinline__ float softplus_f(float x) {
  float ax = fabsf(x);
  float r  = __logf(1.0f + __expf(-ax));
  return (x > 0.0f) ? x + r : r;
}

// ---------------------------------------------------------------------------
// Kernel 1: proj[n][0..3] = emb[n] . {Wu0, Wu1, Wv0, Wv1}
// The only true GEMM in the problem: (50000x32) @ (32x4), done with
// V_WMMA_F32_16X16X4_F32. One wave per 16-row tile, persistent over tiles;
// the 8 B-fragments (tile-invariant) are preloaded once, unconditionally
// (clamped address + v_cndmask zeroing -> no divergent exec branching).
// A layout (16x4 f32):  lane L: m=L&15, kh=L>>4, a={A[m][2kh],A[m][2kh+1]}
// B layout (4x16 f32):  lane L: n=L&15, b={B[2kh][n],B[2kh+1][n]}
// D layout (16x16 f32): VGPR v: row = v + 8*kh, col = n
// ---------------------------------------------------------------------------
__global__ void __launch_bounds__(256)
proj_wmma(const float* __restrict__ emb,
          const float* __restrict__ W,      // (2, 64)
          float* __restrict__ proj,         // (N_NODES, 4)
          int n_waves_total) {
  const int waveId = blockIdx.x * (blockDim.x >> 5) + (threadIdx.x >> 5);
  const int lane   = threadIdx.x & 31;
  const int m      = lane & 15;
  const int kh     = lane >> 4;             // 0 or 1

  // Column n of B (only n<4 non-zero):
  //   n=0 -> Wu[0], n=1 -> Wu[1], n=2 -> Wv[0], n=3 -> Wv[1]
  const int  n    = m;
  const bool bcol = (n < 4);
  const int  nc   = bcol ? n : 3;           // clamp -> always in-bounds load
  const float* wcol = W + (nc & 1) * 64 + (nc >> 1) * 32;

  // Preload the 8 K-step B fragments once (reused across all tiles).
  v2f bv[8];
#pragma unroll
  for (int s = 0; s < 8; ++s) {
    const int k0 = 4 * s + 2 * kh;
    const float wx = wcol[k0];
    const float wy = wcol[k0 + 1];
    bv[s].x = bcol ? wx : 0.0f;             // v_cndmask, no exec branch
    bv[s].y = bcol ? wy : 0.0f;
  }

  // Wave-uniform persistent loop: EXEC stays all-ones around the WMMAs.
  for (int tile = waveId; tile < N_TILES; tile += n_waves_total) {
    const float* aptr = emb + (size_t)(tile * 16 + m) * DIM;
    v8f c = {};
#pragma unroll
    for (int s = 0; s < 8; ++s) {
      const int k0 = 4 * s + 2 * kh;
      v2f a; a.x = aptr[k0]; a.y = aptr[k0 + 1];   // -> global_load_b64
      // 8 args: (neg_a, A, neg_b, B, c_mod, C, reuse_a, reuse_b)
      c = __builtin_amdgcn_wmma_f32_16x16x4_f32(false, a, false, bv[s],
                                                (short)0, c, false, false);
    }
    if (bcol) {
#pragma unroll
      for (int v = 0; v < 8; ++v) {
        proj[(tile * 16 + v + 8 * kh) * 4 + n] = c[v];
      }
    }
  }
}

// ---------------------------------------------------------------------------
// Kernel 2: per-batch scalars (512 threads total; trivial cost)
//   a_src[b] = z_src.Wu[e] + b[e] + alpha[e]*exp(-w_t[e]*td_src/100)
//   a_dst[b] = z_dst.Wv[e] + b[e] + alpha[e]*exp(-w_t[e]*td_dst/100)
//   gconst[b] = z_src.Wu[e] + z_dst.Wv[e] + b[e]
// ---------------------------------------------------------------------------
__global__ void prep_batch(const float* __restrict__ emb,
                           const int*   __restrict__ assoc,
                           const int*   __restrict__ src,
                           const int*   __restrict__ dst,
                           const float* __restrict__ last_update,
                           const float* __restrict__ cur_time,
                           const int*   __restrict__ et,
                           const float* __restrict__ W,
                           const float* __restrict__ bias,
                           const float* __restrict__ psi,
                           const float* __restrict__ alpha,
                           const float* __restrict__ w_t,
                           float* __restrict__ a_src,
                           float* __restrict__ a_dst,
                           float* __restrict__ psi_w,
                           float* __restrict__ inv_w,
                           float* __restrict__ gconst,
                           float* __restrict__ e_f) {
  const int b = blockIdx.x * blockDim.x + threadIdx.x;
  if (b >= BATCH) return;
  const int is = assoc[src[b]];
  const int id = assoc[dst[b]];
  const int e  = (et[b] > 0) ? 1 : 0;
  const float* zs = emb + (size_t)is * DIM;
  const float* zd = emb + (size_t)id * DIM;
  const float* Wu = W + e * 64;
  const float* Wv = Wu + 32;
  float du = 0.0f, dv = 0.0f;
#pragma unroll
  for (int k = 0; k < DIM; ++k) { du += zs[k] * Wu[k]; dv += zd[k] * Wv[k]; }
  const float ct  = cur_time[b];
  const float tds = ct - last_update[is];
  const float tdd = ct - last_update[id];
  const float al  = alpha[e], wt = w_t[e], be = bias[e], ps = psi[e];
  a_src[b]  = du + be + al * __expf(-wt * tds * 0.01f);   // td / 100
  a_dst[b]  = dv + be + al * __expf(-wt * tdd * 0.01f);
  psi_w[b]  = ps;
  inv_w[b]  = 1.0f / (ps + 1e-7f);
  gconst[b] = du + dv + be;
  e_f[b]    = (float)e;
}

// ---------------------------------------------------------------------------
// Kernel 3: the bandwidth-bound fill (2 x 512 x 50000 f32 = 204.8 MB stores,
// ~9us HBM roofline @ 23.3 TB/s). Proj table (800 KB) stays L2-resident;
// outputs streamed with 128-bit non-temporal stores.
//   lambda_src[b,n] = psi_e * softplus((a_src[b] + proj[n][2+e]) * inv_psi)
//   lambda_dst[b,n] = psi_e * softplus((a_dst[b] + proj[n][ e ]) * inv_psi)
// ---------------------------------------------------------------------------
__global__ void __launch_bounds__(256)
fill_lambda(const v4f* __restrict__ proj4,  // N_NODES entries
            const float* __restrict__ a_src,
            const float* __restrict__ a_dst,
            const float* __restrict__ psi_w,
            const float* __restrict__ inv_w,
            const float* __restrict__ e_f,
            float* __restrict__ out) {
  const int b  = blockIdx.y;
  const int g  = blockIdx.x * blockDim.x + threadIdx.x;   // group of 4 nodes
  const int n0 = g * 4;
  if (n0 >= N_NODES) return;                              // N_NODES % 4 == 0
  const float as  = a_src[b], ad = a_dst[b];
  const float ps  = psi_w[b], inv = inv_w[b];
  const bool  e   = e_f[b] > 0.5f;
  v4f rs, rd;
#pragma unroll
  for (int j = 0; j < 4; ++j) {
    const v4f p   = proj4[n0 + j];
    const float pu = e ? p.y : p.x;   // emb.Wu[e]  -> lambda_dst
    const float pv = e ? p.w : p.z;   // emb.Wv[e]  -> lambda_src
    rs[j] = ps * softplus_f((as + pv) * inv);
    rd[j] = ps * softplus_f((ad + pu) * inv);
  }
  float* orow_s = out + (size_t)b * N_NODES + n0;
  float* orow_d = out + (size_t)BATCH * N_NODES + (size_t)b * N_NODES + n0;
  __builtin_nontemporal_store(rs, (v4f*)orow_s);
  __builtin_nontemporal_store(rd, (v4f*)orow_d);
}

// ---------------------------------------------------------------------------
// Kernel 4: return-time prediction. g01 is time-invariant; only the
// alpha*exp(-w_t*t/100) term varies. Inclusive cumsum forces a serial loop;
// one thread per batch element (512 x 1001 steps, negligible).
// ---------------------------------------------------------------------------
__global__ void rtp_kernel(const float* __restrict__ gconst,
                           const float* __restrict__ psi_w,
                           const float* __restrict__ inv_w,
                           const float* __restrict__ e_f,
                           const float* __restrict__ alpha,
                           const float* __restrict__ w_t,
                           float* __restrict__ out_rtp) {
  const int b = blockIdx.x * blockDim.x + threadIdx.x;
  if (b >= BATCH) return;
  const int   e  = (e_f[b] > 0.5f) ? 1 : 0;
  const float gc = gconst[b], ps = psi_w[b], inv = inv_w[b];
  const float al = alpha[e],  wt = w_t[e];
  float integral = 0.0f, acc = 0.0f, ts_prev = 0.0f;
  for (int i = 0; i < NSAMP; ++i) {
    const float td    = (float)i * 0.01f;                 // linspace(0,10,1001)
    const float gg    = gc + al * __expf(-wt * td * 0.01f);
    const float inten = ps * softplus_f(gg * inv);
    integral += 0.01f * inten;                            // inclusive cumsum
    const float ts = td * inten * __expf(-integral);
    if (i > 0) acc += 0.005f * (ts_prev + ts);            // trapezoid
    ts_prev = ts;
  }
  out_rtp[b] = acc;
}

// ---------------------------------------------------------------------------
// Launch
// ---------------------------------------------------------------------------
extern "C" void kernel_launch(void* const* d_in, const int* in_sizes, int n_in,
                              void* d_out, int out_size, void* d_ws,
                              size_t ws_size, hipStream_t stream) {
  (void)in_sizes; (void)n_in; (void)out_size; (void)ws_size;
  const float* emb         = (const float*)d_in[0];
  const int*   assoc       = (const int*)  d_in[1];
  const int*   src         = (const int*)  d_in[2];
  const int*   pos_dst     = (const int*)  d_in[3];
  /* d_in[4] = neg_dst : unused by the reference outputs */
  const float* last_update = (const float*)d_in[5];
  const float* cur_time    = (const float*)d_in[6];
  const int*   et          = (const int*)  d_in[7];
  const float* W           = (const float*)d_in[8];
  const float* bias        = (const float*)d_in[9];
  const float* psi         = (const float*)d_in[10];
  const float* alpha       = (const float*)d_in[11];
  const float* w_t         = (const float*)d_in[12];

  // Workspace layout (floats): proj (N*4) | a_src | a_dst | psi | inv | gc | e
  float* ws     = (float*)d_ws;
  float* proj   = ws;                        // 200000 floats (800 KB)
  float* a_src  = ws + (size_t)N_NODES * 4;
  float* a_dst  = a_src + BATCH;
  float* psi_w  = a_dst + BATCH;
  float* inv_w  = psi_w + BATCH;
  float* gconst = inv_w + BATCH;
  float* e_f    = gconst + BATCH;
  float* out    = (float*)d_out;

  // 1) WMMA projection: 512 persistent waves sweep 3125 tiles (~6 each),
  //    B fragments loaded once per wave.
  const int proj_blocks = 64;                // 64 blocks * 8 waves = 512 waves
  proj_wmma<<<proj_blocks, 256, 0, stream>>>(emb, W, proj, proj_blocks * 8);

  // 2) Per-batch scalars.
  prep_batch<<<(BATCH + 255) / 256, 256, 0, stream>>>(
      emb, assoc, src, pos_dst, last_update, cur_time, et, W, bias, psi, alpha,
      w_t, a_src, a_dst, psi_w, inv_w, gconst, e_f);

  // 3) Bandwidth-bound fill of lambda_src / lambda_dst.
  dim3 grid((N_NODES / 4 + 255) / 256, BATCH);
  fill_lambda<<<grid, 256, 0, stream>>>((const v4f*)proj, a_src, a_dst, psi_w,
                                        inv_w, e_f, out);

  // 4) Return-time integral.
  rtp_kernel<<<(BATCH + 255) / 256, 256, 0, stream>>>(
      gconst, psi_w, inv_w, e_f, alpha, w_t,
      out + 2 * (size_t)BATCH * N_NODES);
}